// Net_75900662055229
// MI455X (gfx1250) — compile-verified
//
#include <hip/hip_runtime.h>

#define DIM 128
#define NNODES 100000
#define NEDGES 3200000

typedef __attribute__((ext_vector_type(2))) float v2f;
typedef __attribute__((ext_vector_type(8))) float v8f;

// ---------------- degree / normalization ----------------

__global__ __launch_bounds__(256) void k_deg_init(float* deg, int n) {
  int i = blockIdx.x * 256 + threadIdx.x;
  if (i < n) deg[i] = 1.0f;  // self-loop
}

__global__ __launch_bounds__(256) void k_deg_count(const int* __restrict__ dst,
                                                   float* deg, int e) {
  int i = blockIdx.x * 256 + threadIdx.x;
  if (i < e) unsafeAtomicAdd(&deg[dst[i]], 1.0f);
}

__global__ __launch_bounds__(256) void k_rsqrt(float* v, int n) {
  int i = blockIdx.x * 256 + threadIdx.x;
  if (i < n) v[i] = 1.0f / sqrtf(v[i]);  // deg >= 1 always
}

// ---------------- dense transform: tmp = H @ W  (fp32 WMMA) ----------------
// One wave per 16-row M-tile, full N=128 strip (8 16x16 accumulators).

__global__ __launch_bounds__(256) void k_gemm(const float* __restrict__ A,
                                              const float* __restrict__ W,
                                              float* __restrict__ C, int M) {
  __shared__ float sW[DIM * DIM];  // 64 KB, whole weight matrix
  for (int i = threadIdx.x; i < DIM * DIM; i += 256) sW[i] = W[i];
  __syncthreads();

  const int wave = threadIdx.x >> 5;
  const int lane = threadIdx.x & 31;
  const int tile_m = blockIdx.x * 8 + wave;
  const int row0 = tile_m * 16;
  if (row0 >= M) return;  // wave-uniform: EXEC stays all-ones for WMMA

  const int r    = lane & 15;         // M within tile (A) / N within tile (B,D)
  const int kh   = (lane >> 4) << 1;  // lane-half selects K pair {0,1} vs {2,3}
  const int half = lane >> 4;         // lane-half selects M vs M+8 for C/D

  v8f acc[8] = {};
  const float* __restrict__ Arow = A + (size_t)(row0 + r) * DIM;

  for (int k0 = 0; k0 < DIM; k0 += 4) {
    v2f a;
    a.x = Arow[k0 + kh];
    a.y = Arow[k0 + kh + 1];
#pragma unroll
    for (int nt = 0; nt < 8; ++nt) {
      const int col = nt * 16 + r;
      v2f b;
      b.x = sW[(k0 + kh) * DIM + col];
      b.y = sW[(k0 + kh + 1) * DIM + col];
      acc[nt] = __builtin_amdgcn_wmma_f32_16x16x4_f32(
          /*neg_a=*/false, a, /*neg_b=*/false, b,
          /*c_mod=*/(short)0, acc[nt], /*reuse_a=*/false, /*reuse_b=*/false);
    }
  }

#pragma unroll
  for (int nt = 0; nt < 8; ++nt) {
#pragma unroll
    for (int rr = 0; rr < 8; ++rr) {
      const int row = row0 + half * 8 + rr;
      C[(size_t)row * DIM + nt * 16 + r] = acc[nt][rr];
    }
  }
}

// ---------------- self-loop + bias init: out = dinv^2 * tmp + b ----------------

__global__ __launch_bounds__(256) void k_agg_init(const float* __restrict__ tmp,
                                                  const float* __restrict__ dinv,
                                                  const float* __restrict__ bias,
                                                  float* __restrict__ out, int n) {
  size_t i = (size_t)blockIdx.x * 256 + threadIdx.x;
  if (i >= (size_t)n * DIM) return;
  int node = (int)(i >> 7);
  int d = (int)(i & (DIM - 1));
  float di = dinv[node];
  out[i] = di * di * tmp[i] + bias[d];
}

// ---------------- edge scatter: one wave per edge, float4 per lane ----------------

__global__ __launch_bounds__(256) void k_scatter(const int* __restrict__ src,
                                                 const int* __restrict__ dst,
                                                 const float* __restrict__ dinv,
                                                 const float* __restrict__ tmp,
                                                 float* __restrict__ out, int e) {
  int edge = blockIdx.x * 8 + (threadIdx.x >> 5);
  int lane = threadIdx.x & 31;
  if (edge >= e) return;
  int s = src[edge];
  int d = dst[edge];
  float w = dinv[s] * dinv[d];
  const float4* tv = (const float4*)(tmp + (size_t)s * DIM);
  float4 v = tv[lane];
  float* o = out + (size_t)d * DIM + lane * 4;
  unsafeAtomicAdd(o + 0, w * v.x);
  unsafeAtomicAdd(o + 1, w * v.y);
  unsafeAtomicAdd(o + 2, w * v.z);
  unsafeAtomicAdd(o + 3, w * v.w);
}

__global__ __launch_bounds__(256) void k_relu(float* h, size_t total) {
  size_t i = (size_t)blockIdx.x * 256 + threadIdx.x;
  if (i < total) h[i] = fmaxf(h[i], 0.0f);
}

// ---------------- final mean over nodes ----------------

__global__ void k_zero(float* o) { o[threadIdx.x] = 0.0f; }

__global__ __launch_bounds__(DIM) void k_reduce(const float* __restrict__ h,
                                                float* __restrict__ out, int n) {
  int d = threadIdx.x;
  int base = blockIdx.x * 256;
  float s = 0.0f;
  for (int i = 0; i < 256; ++i) {
    int node = base + i;
    if (node < n) s += h[(size_t)node * DIM + d];
  }
  unsafeAtomicAdd(&out[d], s * (1.0f / (float)NNODES));
}

// ---------------- driver ----------------

extern "C" void kernel_launch(void* const* d_in, const int* in_sizes, int n_in,
                              void* d_out, int out_size, void* d_ws, size_t ws_size,
                              hipStream_t stream) {
  const float* x  = (const float*)d_in[0];
  const int*   ei = (const int*)d_in[1];
  const float* w1 = (const float*)d_in[2];
  const float* b1 = (const float*)d_in[3];
  const float* w2 = (const float*)d_in[4];
  const float* b2 = (const float*)d_in[5];
  const float* w3 = (const float*)d_in[6];
  const float* b3 = (const float*)d_in[7];
  const int* esrc = ei;            // edge_index[0]
  const int* edst = ei + NEDGES;   // edge_index[1]
  float* out = (float*)d_out;

  char* ws = (char*)d_ws;
  const size_t matB = (size_t)NNODES * DIM * sizeof(float);  // 51.2 MB
  float* tmp  = (float*)(ws);
  float* hA   = (float*)(ws + matB);
  float* hB   = (float*)(ws + 2 * matB);
  float* dinv = (float*)(ws + 3 * matB);

  const int nblk   = (NNODES + 255) / 256;
  const size_t tot = (size_t)NNODES * DIM;
  const int tblk   = (int)((tot + 255) / 256);
  const int gemmg  = (NNODES / 16 + 7) / 8;     // 782 blocks of 8 waves
  const int eblk   = (NEDGES + 7) / 8;          // one edge per wave, 8 waves/block
  const int rblk   = (NNODES + 255) / 256;

  // normalization
  k_deg_init<<<nblk, 256, 0, stream>>>(dinv, NNODES);
  k_deg_count<<<(NEDGES + 255) / 256, 256, 0, stream>>>(edst, dinv, NEDGES);
  k_rsqrt<<<nblk, 256, 0, stream>>>(dinv, NNODES);

  // layer 1
  k_gemm<<<gemmg, 256, 0, stream>>>(x, w1, tmp, NNODES);
  k_agg_init<<<tblk, 256, 0, stream>>>(tmp, dinv, b1, hA, NNODES);
  k_scatter<<<eblk, 256, 0, stream>>>(esrc, edst, dinv, tmp, hA, NEDGES);
  k_relu<<<tblk, 256, 0, stream>>>(hA, tot);

  // layer 2
  k_gemm<<<gemmg, 256, 0, stream>>>(hA, w2, tmp, NNODES);
  k_agg_init<<<tblk, 256, 0, stream>>>(tmp, dinv, b2, hB, NNODES);
  k_scatter<<<eblk, 256, 0, stream>>>(esrc, edst, dinv, tmp, hB, NEDGES);
  k_relu<<<tblk, 256, 0, stream>>>(hB, tot);

  // layer 3 (no relu)
  k_gemm<<<gemmg, 256, 0, stream>>>(hB, w3, tmp, NNODES);
  k_agg_init<<<tblk, 256, 0, stream>>>(tmp, dinv, b3, hA, NNODES);
  k_scatter<<<eblk, 256, 0, stream>>>(esrc, edst, dinv, tmp, hA, NEDGES);

  // mean over nodes -> [1, 128]
  k_zero<<<1, DIM, 0, stream>>>(out);
  k_reduce<<<rblk, DIM, 0, stream>>>(hA, out, NNODES);
}